// PolyGAN_CP_Layer_32787780338354
// MI455X (gfx1250) — compile-verified
//
#include <hip/hip_runtime.h>

// Problem constants from the reference
#define S_TOT  65536   // 256*256
#define RANK   32

typedef __attribute__((ext_vector_type(2))) float v2f;
typedef __attribute__((ext_vector_type(8))) float v8f;

// ---------------------------------------------------------------------------
// Kernel 0: zero the 32-float d1 accumulator in workspace (graph-safe memset)
// ---------------------------------------------------------------------------
__global__ void zero32_kernel(float* __restrict__ ws) {
    if (threadIdx.x < RANK) ws[threadIdx.x] = 0.0f;
}

// ---------------------------------------------------------------------------
// Kernel 1: d1[r] = sum_s z[s] * W1[s,r]   (bandwidth-bound reduction, 8 MB)
// Layout: W1 row-major (s, r), r contiguous. Thread (row,rq) loads float4 of
// ranks rq*4..rq*4+3 for row s -> each wave reads 4 full 128B rows per iter.
// ---------------------------------------------------------------------------
__global__ void reduce_d1_kernel(const float* __restrict__ z,
                                 const float* __restrict__ W1,
                                 float* __restrict__ d1_ws) {
    __shared__ float sred[256][4];

    const int tid = threadIdx.x;
    const int rq  = tid & 7;        // which rank-quad (r = rq*4 .. rq*4+3)
    const int row = tid >> 3;       // 0..31 rows handled per pass
    const int base = blockIdx.x * 256;  // 256 rows per block, 256 blocks

    float a0 = 0.f, a1 = 0.f, a2 = 0.f, a3 = 0.f;
#pragma unroll
    for (int i = 0; i < 8; ++i) {
        const int s  = base + i * 32 + row;
        const float zz = z[s];  // same address across 8 lanes -> broadcast
        const float4 w = *(const float4*)(W1 + (size_t)s * RANK + rq * 4);
        a0 += zz * w.x;
        a1 += zz * w.y;
        a2 += zz * w.z;
        a3 += zz * w.w;
    }

    sred[tid][0] = a0; sred[tid][1] = a1; sred[tid][2] = a2; sred[tid][3] = a3;
    __syncthreads();

    // Tree-reduce across rows (threads with equal rq are congruent mod 8)
#pragma unroll
    for (int stride = 128; stride >= 8; stride >>= 1) {
        if (tid < stride) {
            sred[tid][0] += sred[tid + stride][0];
            sred[tid][1] += sred[tid + stride][1];
            sred[tid][2] += sred[tid + stride][2];
            sred[tid][3] += sred[tid + stride][3];
        }
        __syncthreads();
    }

    if (tid < 8) {
#pragma unroll
        for (int j = 0; j < 4; ++j)
            atomicAdd(&d1_ws[tid * 4 + j], sred[tid][j]);
    }
}

// ---------------------------------------------------------------------------
// Kernel 2: out[s] = b[s] + sum_r W0[s,r] * (2 + d1[r])  via V_WMMA_F32_16X16X4_F32
//
// Per wave: one 16-row tile of W0. Rank-1 trick:
//   A(16x4)[m][k] = v[r0+k]            (replicated over M)
//   B(4x16)[k][n] = W0[s0+n, r0+k]     (tile rows in the N dimension)
//   D[m][n] = sum_k v[r0+k]*W0[s0+n,r0+k]  -> replicated over M
// After 8 accumulating WMMAs (K=32), lane L holds out[s0+(L&15)] in c[0].
//
// ISA 16x4 f32 A layout: VGPR0 = {K=0 | K=2}, VGPR1 = {K=1 | K=3} with the
// K split at lane 16 -> kbase = (lane>>4)*2; B mirrors (4x16, rows across lanes).
// ---------------------------------------------------------------------------
__global__ void matvec_wmma_kernel(const float* __restrict__ W0,
                                   const float* __restrict__ bvec,
                                   const float* __restrict__ d1_ws,
                                   float* __restrict__ out) {
    const int lane   = threadIdx.x & 31;
    const int waveId = (blockIdx.x * blockDim.x + threadIdx.x) >> 5;
    const int s0     = waveId * 16;

    const int n  = lane & 15;         // N index = which of the 16 rows
    const int kb = (lane >> 4) * 2;   // K base for this half-wave

    const float* row = W0 + (size_t)(s0 + n) * RANK;

    v8f c = {};  // zero accumulator
#pragma unroll
    for (int r0 = 0; r0 < RANK; r0 += 4) {
        v2f a, bb;
        a.x  = 2.0f + d1_ws[r0 + kb];
        a.y  = 2.0f + d1_ws[r0 + kb + 1];
        bb.x = row[r0 + kb];
        bb.y = row[r0 + kb + 1];
        // D = A * B + C   (f32 WMMA, 8-arg pattern: neg_a, A, neg_b, B, c_mod, C, reuse_a, reuse_b)
        c = __builtin_amdgcn_wmma_f32_16x16x4_f32(
                false, a, false, bb, (short)0, c, false, false);
    }

    // Result replicated over M: every lane's c[0] is out[s0 + (lane&15)].
    if (lane < 16) {
        out[s0 + n] = c[0] + bvec[s0 + n];
    }
}

// ---------------------------------------------------------------------------
extern "C" void kernel_launch(void* const* d_in, const int* in_sizes, int n_in,
                              void* d_out, int out_size, void* d_ws, size_t ws_size,
                              hipStream_t stream) {
    const float* z = (const float*)d_in[0];        // (S,)
    const float* b = (const float*)d_in[1];        // (S,)
    const float* W = (const float*)d_in[2];        // (9, S, 32) flat
    float* out = (float*)d_out;                    // (S,)
    float* ws  = (float*)d_ws;                     // >= 32 floats

    const float* W0 = W;                            // k = 0
    const float* W1 = W + (size_t)S_TOT * RANK;     // k = 1 (only other live slice)

    zero32_kernel<<<1, 32, 0, stream>>>(ws);
    reduce_d1_kernel<<<256, 256, 0, stream>>>(z, W1, ws);
    // S/16 = 4096 tiles, one wave each -> 4096*32 threads, 256/block -> 512 blocks
    matvec_wmma_kernel<<<512, 256, 0, stream>>>(W0, b, ws, out);
}